// FM_44538810859522
// MI455X (gfx1250) — compile-verified
//
#include <hip/hip_runtime.h>
#include <hip/hip_bf16.h>

typedef __attribute__((ext_vector_type(2))) float v2f;
typedef __attribute__((ext_vector_type(8))) float v8f;

namespace {
constexpr int kEmb             = 16;
constexpr int kBatch           = 2048;
constexpr int kWavesPerBlock   = 8;
constexpr int kSamplesPerWave  = 16;
constexpr int kBlock           = 32 * kWavesPerBlock;                 // 256 threads
constexpr int kSamplesPerBlock = kWavesPerBlock * kSamplesPerWave;    // 128
constexpr int kRowBase         = 2;        // one-hot rows start after 2 numeric rows
constexpr int kOff1 = 100000;              // cumsum of CAT_SIZES (exclusive)
constexpr int kOff2 = 150000;
constexpr int kOff3 = 150020;
}

__global__ __launch_bounds__(kBlock) void fm_forward_kernel(
    const long long* __restrict__ x,        // (2048, 6) int64
    const float*     __restrict__ v,        // (150122, 16) f32
    const float*     __restrict__ bias_user,
    const float*     __restrict__ bias_item,
    const float*     __restrict__ bias_genre,
    const float*     __restrict__ bias_year,
    float*           __restrict__ out)      // (2048,) f32
{
  __shared__ float S[kWavesPerBlock][kSamplesPerWave * 17];  // padded 16x16 tiles

  const int tid  = threadIdx.x;
  const int wave = tid >> 5;
  const int lane = tid & 31;
  const int m    = lane & 15;   // sample within wave tile
  const int h    = lane >> 4;   // dim half: 0 -> dims 0..7, 1 -> dims 8..15

  const int sample = (blockIdx.x * kWavesPerBlock + wave) * kSamplesPerWave + m;

  const long long* xr = x + (long long)sample * 6;
  const float w0 = (float)xr[0];
  const float w1 = (float)xr[1];
  const int c0 = (int)xr[2];
  const int c1 = (int)xr[3];
  const int c2 = (int)xr[4];
  const int c3 = (int)xr[5];

  // s[d] over this lane's 8 dims; q = sum_d x^2 * v^2 (partial over 8 dims)
  float s[8];
  float q = 0.0f;
  {
    const float* p0 = v + 0 * kEmb + h * 8;
    const float* p1 = v + 1 * kEmb + h * 8;
    #pragma unroll
    for (int d = 0; d < 8; ++d) {
      const float t0 = w0 * p0[d];
      const float t1 = w1 * p1[d];
      s[d] = t0 + t1;
      q += t0 * t0 + t1 * t1;
    }
  }
  const int rows[4] = { kRowBase + c0, kRowBase + kOff1 + c1,
                        kRowBase + kOff2 + c2, kRowBase + kOff3 + c3 };
  #pragma unroll
  for (int f = 0; f < 4; ++f) {
    const float* p = v + (long long)rows[f] * kEmb + h * 8;
    #pragma unroll
    for (int d = 0; d < 8; ++d) {
      const float val = p[d];     // one-hot weight is 1.0
      s[d] += val;
      q += val * val;
    }
  }

  // Linear bias term (both half-lanes of a sample compute the same value).
  const float bias = bias_user[c0] + bias_item[c1] + bias_genre[c2] + bias_year[c3];

  // Combine the two dim-halves of q (lanes m and m+16 hold the halves).
  q += __shfl_xor(q, 16, 32);

  // Stage this wave's 16x16 sum-matrix tile in LDS (rows padded to 17 words).
  float* St = &S[wave][0];
  #pragma unroll
  for (int d = 0; d < 8; ++d) St[m * 17 + h * 8 + d] = s[d];
  __syncthreads();

  // Gram diagonal via V_WMMA_F32_16X16X4_F32: D = S * S^T, K=16 as 4 chunks of 4.
  // A 16x4 f32 layout (ISA 7.12.2): lanes 0-15 -> M=lane, VGPR0=K0, VGPR1=K1;
  //                                 lanes 16-31 -> M=lane-16, VGPR0=K2, VGPR1=K3.
  // B 4x16 layout mirrors C/D row-striping; for B = S[:,chunk]^T the per-lane
  // operand values coincide with A's, so the same registers feed both inputs.
  v8f acc = {};
  #pragma unroll
  for (int i = 0; i < 4; ++i) {
    v2f a;
    a.x = St[m * 17 + 4 * i + 2 * h];
    a.y = St[m * 17 + 4 * i + 2 * h + 1];
    acc = __builtin_amdgcn_wmma_f32_16x16x4_f32(
        /*neg_a=*/false, a, /*neg_b=*/false, a,
        /*c_mod=*/(short)0, acc, /*reuse_a=*/false, /*reuse_b=*/false);
  }

  // D[m][m] location (16x16 f32 C/D layout): m<8 -> lane m, VGPR m;
  //                                          m>=8 -> lane m+16, VGPR m-8.
  const bool valid = (lane < 8) || (lane >= 24);
  const int  idx   = (lane < 8) ? lane : (lane - 24);
  float norm2 = 0.0f;
  #pragma unroll
  for (int r = 0; r < 8; ++r) norm2 += (idx == r) ? (float)acc[r] : 0.0f;

  if (valid) {
    const float y = bias + 0.5f * (norm2 - q);
    out[sample] = 1.0f / (1.0f + expf(-y));   // sigmoid
  }
}

extern "C" void kernel_launch(void* const* d_in, const int* in_sizes, int n_in,
                              void* d_out, int out_size, void* d_ws, size_t ws_size,
                              hipStream_t stream) {
  const long long* x  = (const long long*)d_in[0];   // int64 features
  const float*     v  = (const float*)d_in[1];
  const float*     bu = (const float*)d_in[2];
  const float*     bi = (const float*)d_in[3];
  const float*     bg = (const float*)d_in[4];
  const float*     by = (const float*)d_in[5];
  float*           out = (float*)d_out;
  (void)in_sizes; (void)n_in; (void)out_size; (void)d_ws; (void)ws_size;

  dim3 grid(kBatch / kSamplesPerBlock);   // 16 blocks
  dim3 block(kBlock);                     // 256 threads = 8 waves
  hipLaunchKernelGGL(fm_forward_kernel, grid, block, 0, stream,
                     x, v, bu, bi, bg, by, out);
}